// Group_85942295593336
// MI455X (gfx1250) — compile-verified
//
#include <hip/hip_runtime.h>

typedef float v2f __attribute__((ext_vector_type(2)));
typedef float v8f __attribute__((ext_vector_type(8)));

constexpr int NB = 32;     // batches
constexpr int NP = 8192;   // points per batch
constexpr int NG = 512;    // groups (FPS samples)
constexpr int GS = 32;     // neighbors per group
constexpr int NTILES = NP / 16;
constexpr int PF = 8;      // prefetch distance (tiles)

// ---------------------------------------------------------------------------
// Kernel 1: furthest point sampling. One block per batch, 1024 threads
// (32 wave32s). Each thread caches 8 points + running min-distance in
// registers; per step: distance update + argmax via wave shuffles + tiny LDS.
// ---------------------------------------------------------------------------
__global__ __launch_bounds__(1024)
void fps_kernel(const float* __restrict__ xyz,
                float* __restrict__ centers,
                float* __restrict__ out_center) {
  const int b   = blockIdx.x;
  const int tid = threadIdx.x;

  __shared__ float s_c[3];
  __shared__ float s_bd[32];
  __shared__ int   s_bi[32];
  __shared__ int   s_next;

  float px[8], py[8], pz[8], dist[8];
  const float* base = xyz + (size_t)b * NP * 6;
#pragma unroll
  for (int k = 0; k < 8; ++k) {
    const float* p = base + (size_t)(k * 1024 + tid) * 6;
    px[k] = p[0]; py[k] = p[1]; pz[k] = p[2];
    dist[k] = 1.0e10f;
  }

  int cur = 0;
  for (int it = 0; it < NG; ++it) {
    if (tid == 0) {
      const float* c = base + (size_t)cur * 6;
      float cx = c[0], cy = c[1], cz = c[2];
      s_c[0] = cx; s_c[1] = cy; s_c[2] = cz;
      float* cc = centers    + ((size_t)b * NG + it) * 3;
      float* oc = out_center + ((size_t)b * NG + it) * 3;
      cc[0] = cx; cc[1] = cy; cc[2] = cz;
      oc[0] = cx; oc[1] = cy; oc[2] = cz;
    }
    __syncthreads();
    const float cx = s_c[0], cy = s_c[1], cz = s_c[2];

    float bd = -1.0f; int bi = 0x7fffffff;
#pragma unroll
    for (int k = 0; k < 8; ++k) {
      float dx = px[k] - cx, dy = py[k] - cy, dz = pz[k] - cz;
      float d  = dx * dx + dy * dy + dz * dz;
      d = fminf(dist[k], d);
      dist[k] = d;
      int idx = k * 1024 + tid;
      if (d > bd || (d == bd && idx < bi)) { bd = d; bi = idx; }
    }
    // wave32 reduction (argmax, lowest index on ties)
#pragma unroll
    for (int off = 16; off > 0; off >>= 1) {
      float od = __shfl_down(bd, off, 32);
      int   oi = __shfl_down(bi, off, 32);
      if (od > bd || (od == bd && oi < bi)) { bd = od; bi = oi; }
    }
    if ((tid & 31) == 0) { s_bd[tid >> 5] = bd; s_bi[tid >> 5] = bi; }
    __syncthreads();
    if (tid < 32) {
      bd = s_bd[tid]; bi = s_bi[tid];
#pragma unroll
      for (int off = 16; off > 0; off >>= 1) {
        float od = __shfl_down(bd, off, 32);
        int   oi = __shfl_down(bi, off, 32);
        if (od > bd || (od == bd && oi < bi)) { bd = od; bi = oi; }
      }
      if (tid == 0) s_next = bi;
    }
    __syncthreads();
    cur = s_next;
  }
}

// ---------------------------------------------------------------------------
// Kernel 2: kNN via WMMA. One wave32 per block; block owns a tile of 16
// groups for one batch. A (16x4 f32: xyz padded with 0) stays resident in
// 2 VGPRs; loop over 512 point-tiles of 16, each one V_WMMA_F32_16X16X4_F32.
// Next tile's B coords are loaded unconditionally (clamped index); since the
// block is a SINGLE wave, no barriers are needed inside the loop (same-wave
// DS ops complete in order), so the load overlaps WMMA + the whole selection
// scan. WGP-scope global_prefetch_b8 (inline asm) pulls lines into the WGP
// cache PF tiles ahead. Selection scans the 16x16 d2 tile as float4 quads
// (ds_load_b128) with a threshold fast-path; sorted top-32 lists in LDS.
// ---------------------------------------------------------------------------
__global__ __launch_bounds__(32)
void knn_kernel(const float* __restrict__ xyz,
                const float* __restrict__ centers,
                int* __restrict__ knn) {
  const int b    = blockIdx.y;      // batch
  const int mt   = blockIdx.x;      // group-tile (16 groups)
  const int lane = threadIdx.x;     // 0..31
  const int hi   = lane >> 4;       // half-wave select
  const int lm   = lane & 15;

  __shared__ float s_cn[16];
  __shared__ alignas(16) float s_tile[16][16];
  __shared__ float s_ld[16][GS];
  __shared__ int   s_li[16][GS];

  // --- A tile: centers. ISA 16x4 f32 A layout:
  // lanes 0-15: (K=0,K=1) = (x,y); lanes 16-31: (K=2,K=3) = (z,0)
  const float* cm = centers + ((size_t)b * NG + mt * 16 + lm) * 3;
  const float cxx = cm[0], cyy = cm[1], czz = cm[2];
  s_cn[lm] = cxx * cxx + cyy * cyy + czz * czz;   // both halves write same value
  v2f a;
  if (hi == 0) { a.x = cxx; a.y = cyy; } else { a.x = czz; a.y = 0.0f; }

  // init per-group sorted lists
  if (hi == 0) {
#pragma unroll
    for (int j = 0; j < GS; ++j) { s_ld[lm][j] = 3.0e38f; s_li[lm][j] = 0; }
  }
  __syncthreads();   // once, before the loop (cheap; single wave)

  float cn[8];
#pragma unroll
  for (int r = 0; r < 8; ++r) cn[r] = s_cn[r + 8 * hi];

  const float* pbase = xyz + (size_t)b * NP * 6;

  // prologue: tile 0 coords in registers
  float x = pbase[(size_t)lm * 6 + 0];
  float y = pbase[(size_t)lm * 6 + 1];
  float z = pbase[(size_t)lm * 6 + 2];

  for (int nt = 0; nt < NTILES; ++nt) {
    // WGP-scope prefetch PF tiles ahead (default scope pulls into WGP$)
    if (nt + PF < NTILES) {
      const void* pf = (const void*)(pbase + (size_t)((nt + PF) * 16 + lm) * 6);
      asm volatile("global_prefetch_b8 %0, off" :: "v"(pf));
    }

    // software pipeline: unconditional next-tile load (clamped index);
    // flies under the WMMA + selection below (no in-loop barrier)
    const int nn = (nt + 1 < NTILES) ? (nt + 1) : (NTILES - 1);
    const float* pnext = pbase + (size_t)(nn * 16 + lm) * 6;
    const float nx = pnext[0];
    const float ny = pnext[1];
    const float nz = pnext[2];

    const float pn2 = x * x + y * y + z * z;
    v2f bb;
    if (hi == 0) { bb.x = x; bb.y = y; } else { bb.x = z; bb.y = 0.0f; }

    v8f acc = {};
    acc = __builtin_amdgcn_wmma_f32_16x16x4_f32(
        /*neg_a=*/false, a, /*neg_b=*/false, bb,
        /*c_mod=*/(short)0, acc, /*reuse_a=*/false, /*reuse_b=*/false);

    // d2 tile -> LDS. Lane holds rows M=r+8*hi, column N=lm.
    // Single wave: same-wave DS ops are in-order -> no barrier needed.
#pragma unroll
    for (int r = 0; r < 8; ++r)
      s_tile[r + 8 * hi][lm] = cn[r] + pn2 - 2.0f * acc[r];

    // streaming top-32 insert (lanes 0-15, one group each). Quad scan with
    // threshold fast-path; candidates ascend in point index -> stable order.
    if (hi == 0) {
      float worst = s_ld[lm][GS - 1];
      const float4* row = (const float4*)&s_tile[lm][0];
      for (int q = 0; q < 4; ++q) {
        float4 d4 = row[q];
        if (d4.x < worst || d4.y < worst || d4.z < worst || d4.w < worst) {
          float dv[4] = {d4.x, d4.y, d4.z, d4.w};
#pragma unroll 1
          for (int ii = 0; ii < 4; ++ii) {
            float d = dv[ii];
            if (d < worst) {
              int idx = nt * 16 + q * 4 + ii;
              int pos = GS - 1;
              while (pos > 0 && s_ld[lm][pos - 1] > d) {
                s_ld[lm][pos] = s_ld[lm][pos - 1];
                s_li[lm][pos] = s_li[lm][pos - 1];
                --pos;
              }
              s_ld[lm][pos] = d;
              s_li[lm][pos] = idx;
              worst = s_ld[lm][GS - 1];
            }
          }
        }
      }
    }

    x = nx; y = ny; z = nz;
  }

  // write knn indices for the 16 owned groups (same-wave LDS, in order)
  if (hi == 0) {
    int* kout = knn + ((size_t)b * NG + mt * 16 + lm) * GS;
#pragma unroll
    for (int j = 0; j < GS; ++j) kout[j] = s_li[lm][j];
  }
}

// ---------------------------------------------------------------------------
// Kernel 3: gather neighbors, subtract center from xyz part, pass features.
// One thread per (b, g, j).
// ---------------------------------------------------------------------------
__global__ __launch_bounds__(256)
void gather_kernel(const float* __restrict__ xyz,
                   const float* __restrict__ centers,
                   const int* __restrict__ knn,
                   float* __restrict__ out) {
  const int t = blockIdx.x * 256 + threadIdx.x;   // 0 .. NB*NG*GS-1
  if (t >= NB * NG * GS) return;
  const int bg = t >> 5;            // b*NG + g
  const int b  = t >> 14;           // NG*GS = 16384
  const int idx = knn[t];
  const float* p = xyz + ((size_t)b * NP + idx) * 6;
  const float* c = centers + (size_t)bg * 3;
  float* o = out + (size_t)t * 6;
  o[0] = p[0] - c[0];
  o[1] = p[1] - c[1];
  o[2] = p[2] - c[2];
  o[3] = p[3];
  o[4] = p[4];
  o[5] = p[5];
}

// ---------------------------------------------------------------------------
extern "C" void kernel_launch(void* const* d_in, const int* in_sizes, int n_in,
                              void* d_out, int out_size, void* d_ws, size_t ws_size,
                              hipStream_t stream) {
  const float* xyz = (const float*)d_in[0];
  float* out        = (float*)d_out;
  float* out_center = out + (size_t)NB * NG * GS * 6;   // neighborhood | center

  char*  ws      = (char*)d_ws;
  float* centers = (float*)ws;                                   // NB*NG*3 floats
  int*   knn     = (int*)(ws + sizeof(float) * (size_t)NB * NG * 3);  // NB*NG*GS ints

  fps_kernel<<<NB, 1024, 0, stream>>>(xyz, centers, out_center);

  dim3 g2(NG / 16, NB);
  knn_kernel<<<g2, 32, 0, stream>>>(xyz, centers, knn);

  const int total = NB * NG * GS;
  gather_kernel<<<(total + 255) / 256, 256, 0, stream>>>(xyz, centers, knn, out);
}